// VectorQuantizerEMA_23295902613869
// MI455X (gfx1250) — compile-verified
//
#include <hip/hip_runtime.h>
#include <hip/hip_bf16.h>

// ---------------------------------------------------------------------------
// VQ-VAE EMA vector quantizer for MI455X (gfx1250), fp32 WMMA path.
//   z          : [65536, 512] f32
//   embedding  : [1024, 512]  f32
//   outputs    : quantized(33554432) | indices(65536) | new_embedding(524288)
//                | new_cluster_size(1024) | new_embed_avg(524288)
// Main kernel: 2x2 register-blocked WMMA (32 tokens x 32 codes per wave) so
// both LDS-A and global-B traffic sit at 128 B per wmma — balanced against
// the V_WMMA_F32_16X16X4_F32 issue rate.
// ---------------------------------------------------------------------------

typedef float v2f __attribute__((ext_vector_type(2)));
typedef float v8f __attribute__((ext_vector_type(8)));

#define DIMS       512
#define KCODES     1024
#define NTOK       65536
#define TOK_PER_WG 64
#define LDS_PITCH  516  // 516 % 64 == 4 -> conflict-free banked A-fragment reads

// --------------------------- e_sq precompute -------------------------------
__global__ __launch_bounds__(128) void esq_kernel(const float* __restrict__ emb,
                                                  float* __restrict__ esq) {
  const int k   = blockIdx.x;
  const int tid = threadIdx.x;
  float s = 0.f;
#pragma unroll
  for (int c = tid; c < DIMS; c += 128) {
    float v = emb[k * DIMS + c];
    s += v * v;
  }
#pragma unroll
  for (int m = 16; m >= 1; m >>= 1) s += __shfl_xor(s, m, 32);
  __shared__ float red[4];
  if ((tid & 31) == 0) red[tid >> 5] = s;
  __syncthreads();
  if (tid == 0) esq[k] = red[0] + red[1] + red[2] + red[3];
}

// ------------------- fused distance GEMM + row argmin ----------------------
// WG = 64 threads = 2 waves; each wave owns 32 tokens (2 M-subtiles of 16).
// Codes processed 32 at a time (2 N-subtiles). 2x2 accumulator blocking:
// every A fragment (LDS) and B fragment (global/L2) feeds two wmmas.
__global__ __launch_bounds__(64) void vq_argmin_kernel(
    const float* __restrict__ z, const float* __restrict__ emb,
    const float* __restrict__ esq, int* __restrict__ idx_i,
    float* __restrict__ idx_f) {
  __shared__ float zt[TOK_PER_WG * LDS_PITCH];

  const int tid  = threadIdx.x;
  const int tok0 = blockIdx.x * TOK_PER_WG;

  // Stage 64x512 z-tile into LDS (float4 global loads, b128 LDS stores).
  for (int i = tid * 4; i < TOK_PER_WG * DIMS; i += 64 * 4) {
    const int r = i >> 9;  // / 512
    const int c = i & 511;
    *reinterpret_cast<float4*>(&zt[r * LDS_PITCH + c]) =
        *reinterpret_cast<const float4*>(z + (size_t)(tok0 + r) * DIMS + c);
  }
  __syncthreads();

  const int wave = tid >> 5;           // 0 or 1
  const int lane = tid & 31;
  const int l16  = lane & 15;
  const int lh2  = (lane >> 4) << 1;   // 0 or 2: K-offset of this lane half
  const int mrow = wave << 5;          // this wave's 32-token block

  const float* aBase0 = &zt[(mrow + l16) * LDS_PITCH + lh2];        // M-subtile 0
  const float* aBase1 = &zt[(mrow + 16 + l16) * LDS_PITCH + lh2];   // M-subtile 1

  float bestv[16];
  int   besti[16];
#pragma unroll
  for (int j = 0; j < 16; ++j) { bestv[j] = 3.4e38f; besti[j] = 0; }

  // 32 iterations over 32-code groups (2 N-subtiles each).
  for (int t = 0; t < 32; ++t) {
    const int n0 = t * 32;
    const float* b0 = emb + (size_t)(n0 + l16) * DIMS + lh2;        // N-subtile 0
    const float* b1 = emb + (size_t)(n0 + 16 + l16) * DIMS + lh2;   // N-subtile 1
    if (t < 31) {  // hint next code group toward the caches
      __builtin_prefetch(b0 + 32 * DIMS, 0, 1);
      __builtin_prefetch(b1 + 32 * DIMS, 0, 1);
    }

    v8f acc00 = {}, acc01 = {}, acc10 = {}, acc11 = {};
#pragma unroll 4
    for (int d = 0; d < DIMS; d += 4) {
      const v2f a0 = *reinterpret_cast<const v2f*>(aBase0 + d);  // LDS
      const v2f a1 = *reinterpret_cast<const v2f*>(aBase1 + d);  // LDS
      const v2f q0 = *reinterpret_cast<const v2f*>(b0 + d);      // global/L2
      const v2f q1 = *reinterpret_cast<const v2f*>(b1 + d);      // global/L2
      acc00 = __builtin_amdgcn_wmma_f32_16x16x4_f32(false, a0, false, q0,
                                                    (short)0, acc00, false, false);
      acc01 = __builtin_amdgcn_wmma_f32_16x16x4_f32(false, a0, false, q1,
                                                    (short)0, acc01, false, false);
      acc10 = __builtin_amdgcn_wmma_f32_16x16x4_f32(false, a1, false, q0,
                                                    (short)0, acc10, false, false);
      acc11 = __builtin_amdgcn_wmma_f32_16x16x4_f32(false, a1, false, q1,
                                                    (short)0, acc11, false, false);
    }

    // Fold into running per-lane argmin.
    // dist = ||e||^2 - 2<z,e>  (||z||^2 is constant per row -> same argmin).
    const float en0 = esq[n0 + l16];
    const float en1 = esq[n0 + 16 + l16];
#pragma unroll
    for (int j = 0; j < 8; ++j) {
      // N-subtile 0 first (lower code index), then N-subtile 1: codes ascend.
      float dv = en0 - 2.0f * acc00[j];
      if (dv < bestv[j]) { bestv[j] = dv; besti[j] = n0 + l16; }
      dv = en1 - 2.0f * acc01[j];
      if (dv < bestv[j]) { bestv[j] = dv; besti[j] = n0 + 16 + l16; }
      dv = en0 - 2.0f * acc10[j];
      if (dv < bestv[8 + j]) { bestv[8 + j] = dv; besti[8 + j] = n0 + l16; }
      dv = en1 - 2.0f * acc11[j];
      if (dv < bestv[8 + j]) { bestv[8 + j] = dv; besti[8 + j] = n0 + 16 + l16; }
    }
  }

  // Cross-lane argmin over the 16 lanes of each half (same token per half).
#pragma unroll
  for (int s = 0; s < 16; ++s) {
    const int mi = s >> 3;     // M-subtile
    const int j  = s & 7;      // accumulator VGPR index
    float bv = bestv[s];
    int   bi = besti[s];
#pragma unroll
    for (int m = 1; m <= 8; m <<= 1) {
      const float ov = __shfl_xor(bv, m, 32);
      const int   oi = __shfl_xor(bi, m, 32);
      if (ov < bv || (ov == bv && oi < bi)) { bv = ov; bi = oi; }
    }
    if (l16 == 0) {
      const int tok = tok0 + mrow + (mi << 4) + j + ((lane >> 4) << 3);
      idx_i[tok] = bi;
      idx_f[tok] = (float)bi;
    }
  }
}

// ----------------------- gather quantized vectors --------------------------
__global__ __launch_bounds__(128) void gather_kernel(const float* __restrict__ emb,
                                                     const int* __restrict__ idx,
                                                     float* __restrict__ outq) {
  const int n    = blockIdx.x;
  const int code = idx[n];
  const int c    = threadIdx.x * 4;
  const float4 v = *reinterpret_cast<const float4*>(emb + (size_t)code * DIMS + c);
  *reinterpret_cast<float4*>(outq + (size_t)n * DIMS + c) = v;
}

// ------------------ histogram + segment-sum scatter-add --------------------
__global__ __launch_bounds__(128) void scatter_kernel(const float* __restrict__ z,
                                                      const int* __restrict__ idx,
                                                      float* __restrict__ counts,
                                                      float* __restrict__ esum) {
  const int n    = blockIdx.x;
  const int code = idx[n];
  if (threadIdx.x == 0) atomicAdd(&counts[code], 1.0f);
  const int c    = threadIdx.x * 4;
  const float4 v = *reinterpret_cast<const float4*>(z + (size_t)n * DIMS + c);
  float* dst = esum + (size_t)code * DIMS + c;
  atomicAdd(dst + 0, v.x);
  atomicAdd(dst + 1, v.y);
  atomicAdd(dst + 2, v.z);
  atomicAdd(dst + 3, v.w);
}

// ------------- new_cluster_size + total count reduction --------------------
__global__ __launch_bounds__(1024) void newcs_kernel(const float* __restrict__ cs,
                                                     const float* __restrict__ counts,
                                                     float* __restrict__ out_ncs,
                                                     float* __restrict__ nsum) {
  const int k = threadIdx.x;  // 1024 threads, one block
  const float v = 0.99f * cs[k] + 0.01f * counts[k];
  out_ncs[k] = v;
  float s = v;
#pragma unroll
  for (int m = 16; m >= 1; m >>= 1) s += __shfl_xor(s, m, 32);
  __shared__ float red[32];
  if ((k & 31) == 0) red[k >> 5] = s;
  __syncthreads();
  if (k < 32) {
    float t = red[k];
#pragma unroll
    for (int m = 16; m >= 1; m >>= 1) t += __shfl_xor(t, m, 32);
    if (k == 0) nsum[0] = t;
  }
}

// ------------- EMA embed_avg update + smoothed normalization ---------------
__global__ __launch_bounds__(128) void finalize_kernel(
    const float* __restrict__ ea, const float* __restrict__ esum,
    const float* __restrict__ ncs, const float* __restrict__ nsum,
    float* __restrict__ out_emb, float* __restrict__ out_nea) {
  const int k = blockIdx.x;
  const float n  = nsum[0];
  const float sm = (ncs[k] + 1e-5f) / (n + 1024.0f * 1e-5f) * n;
  const int c = threadIdx.x * 4;
  const size_t off = (size_t)k * DIMS + c;
  const float4 a = *reinterpret_cast<const float4*>(ea + off);
  const float4 s = *reinterpret_cast<const float4*>(esum + off);
  float4 v;
  v.x = 0.99f * a.x + 0.01f * s.x;
  v.y = 0.99f * a.y + 0.01f * s.y;
  v.z = 0.99f * a.z + 0.01f * s.z;
  v.w = 0.99f * a.w + 0.01f * s.w;
  *reinterpret_cast<float4*>(out_nea + off) = v;
  float4 e;
  e.x = v.x / sm;
  e.y = v.y / sm;
  e.z = v.z / sm;
  e.w = v.w / sm;
  *reinterpret_cast<float4*>(out_emb + off) = e;
}

// ---------------------------------------------------------------------------
extern "C" void kernel_launch(void* const* d_in, const int* in_sizes, int n_in,
                              void* d_out, int out_size, void* d_ws, size_t ws_size,
                              hipStream_t stream) {
  (void)in_sizes; (void)n_in; (void)out_size; (void)ws_size;

  const float* z   = (const float*)d_in[0];  // [65536, 512]
  const float* emb = (const float*)d_in[1];  // [1024, 512]
  const float* cs  = (const float*)d_in[2];  // [1024]
  const float* ea  = (const float*)d_in[3];  // [1024, 512]

  float* out      = (float*)d_out;
  float* out_q    = out;                 // 33,554,432 (quantized_st == quantized)
  float* out_idx  = out + 33554432;      //     65,536 (indices as float)
  float* out_emb  = out + 33619968;      //    524,288 (new_embedding)
  float* out_ncs  = out + 34144256;      //      1,024 (new_cluster_size)
  float* out_nea  = out + 34145280;      //    524,288 (new_embed_avg)

  char*  ws     = (char*)d_ws;
  int*   w_idx  = (int*)(ws);                 // 262,144 B
  float* w_esq  = (float*)(ws + 262144);      //   4,096 B
  float* w_cnt  = (float*)(ws + 266240);      //   4,096 B
  float* w_n    = (float*)(ws + 270336);      //      16 B
  float* w_esum = (float*)(ws + 270352);      // 2,097,152 B

  hipMemsetAsync(w_cnt, 0, KCODES * sizeof(float), stream);
  hipMemsetAsync(w_esum, 0, (size_t)KCODES * DIMS * sizeof(float), stream);

  esq_kernel<<<KCODES, 128, 0, stream>>>(emb, w_esq);
  vq_argmin_kernel<<<NTOK / TOK_PER_WG, 64, 0, stream>>>(z, emb, w_esq, w_idx,
                                                         out_idx);
  gather_kernel<<<NTOK, 128, 0, stream>>>(emb, w_idx, out_q);
  scatter_kernel<<<NTOK, 128, 0, stream>>>(z, w_idx, w_cnt, w_esum);
  newcs_kernel<<<1, 1024, 0, stream>>>(cs, w_cnt, out_ncs, w_n);
  finalize_kernel<<<KCODES, 128, 0, stream>>>(ea, w_esum, out_ncs, w_n, out_emb,
                                              out_nea);
}